// EVI_Relu_19353122636473
// MI455X (gfx1250) — compile-verified
//
#include <hip/hip_runtime.h>
#include <hip/hip_bf16.h>

typedef __attribute__((ext_vector_type(2))) float v2f;
typedef __attribute__((ext_vector_type(8))) float v8f;

#define SELU_ALPHA 1.6732632423543772f
#define SELU_SCALE 1.0507009873554805f

// Kernel 1: tiny elementwise pass over mu (B*D elems).
//   mu_g[i] = SCALE * (mu>0 ? mu : ALPHA*(exp(mu)-1))   -> d_out[0 .. BD)
//   g[i]    = SCALE * (mu>0 ? 1  : ALPHA*exp(mu))       -> d_ws (reused by kernel 2)
__global__ void selu_fwd_g_kernel(const float* __restrict__ mu,
                                  float* __restrict__ mu_g,
                                  float* __restrict__ g,
                                  int n) {
    int i = blockIdx.x * blockDim.x + threadIdx.x;
    if (i >= n) return;
    float m = mu[i];
    float e = expf(m);
    bool pos = m > 0.0f;
    mu_g[i] = SELU_SCALE * (pos ? m : SELU_ALPHA * (e - 1.0f));
    g[i]    = SELU_SCALE * (pos ? 1.0f : SELU_ALPHA * e);
}

// Kernel 2: bandwidth-bound streaming of sigma (512 MiB in, 512 MiB out).
// One wave32 per 16x16 (i,j) tile. The rank-1 outer product g_i * g_j^T is one
// V_WMMA_F32_16X16X4_F32 (A: g_i in K=0 column; B: g_j in K=0 row; K=1..3 zero;
// C=0). The D-matrix lands in the ISA 16x16 f32 C/D layout (VGPR r <-> M=r /
// M=8+r; lane <-> N), which we walk with 8 coalesced non-temporal b32 loads and
// stores. D is a template parameter so every row step becomes an IOFFSET
// immediate (r*4*DD bytes, fits signed 24-bit) — one base address, no per-row
// v_add_nc_u64 chains, tight s_clause groups.
template <int DD>
__global__ void __launch_bounds__(256)
sigma_outer_scale_kernel(const float* __restrict__ sigma,
                         const float* __restrict__ g,
                         float* __restrict__ out,
                         int total_tiles) {
    constexpr int NJ = DD >> 4;            // tiles per row
    constexpr int TPB = NJ * NJ;           // tiles per sample

    const int lane = threadIdx.x & 31;
    const int wid  = blockIdx.x * (blockDim.x >> 5) + (threadIdx.x >> 5);
    if (wid >= total_tiles) return;        // wave-uniform: EXEC stays all-1s below

    const int b  = wid / TPB;
    const int t  = wid - b * TPB;
    const int ti = t / NJ;                 // constexpr divisor -> shift
    const int tj = t - ti * NJ;
    const int i0 = ti << 4;
    const int j0 = tj << 4;

    const int  l16 = lane & 15;
    const bool lo  = lane < 16;

    // g row slices for this tile (g is tiny and hot in cache -> regular loads)
    const float* gb = g + (size_t)b * DD;
    const float giv = gb[i0 + l16];
    const float gjv = gb[j0 + l16];

    // A (16x4 f32): VGPR0 lanes0-15 = K=0 column = g_i[M]; all other slots 0.
    // B (4x16 f32): VGPR0 lanes0-15 = K=0 row    = g_j[N]; all other slots 0.
    v2f a  = { lo ? giv : 0.0f, 0.0f };
    v2f bm = { lo ? gjv : 0.0f, 0.0f };
    v8f c  = {};
    c = __builtin_amdgcn_wmma_f32_16x16x4_f32(
            /*neg_a=*/false, a, /*neg_b=*/false, bm,
            /*c_mod=*/(short)0, c, /*reuse_a=*/false, /*reuse_b=*/false);

    // C/D layout walk: VGPR r -> row i0 + rbase + r, lane -> col j0 + (lane&15)
    const int rbase = lo ? 0 : 8;
    const size_t off = (size_t)b * DD * DD + (size_t)(i0 + rbase) * DD + (size_t)(j0 + l16);
    const float* sp = sigma + off;
    float*       op = out + off;

#pragma unroll
    for (int r = 0; r < 8; ++r) {
        float s = __builtin_nontemporal_load(sp + (size_t)r * DD);  // immediate offset, th:NT
        __builtin_nontemporal_store(s * c[r], op + (size_t)r * DD); // immediate offset, th:NT
    }
}

// Generic-D fallback (runtime stride), same algorithm.
__global__ void __launch_bounds__(256)
sigma_outer_scale_generic(const float* __restrict__ sigma,
                          const float* __restrict__ g,
                          float* __restrict__ out,
                          int D, int nj, int tiles_per_b, int total_tiles) {
    const int lane = threadIdx.x & 31;
    const int wid  = blockIdx.x * (blockDim.x >> 5) + (threadIdx.x >> 5);
    if (wid >= total_tiles) return;

    const int b  = wid / tiles_per_b;
    const int t  = wid - b * tiles_per_b;
    const int ti = t / nj;
    const int tj = t - ti * nj;
    const int i0 = ti << 4;
    const int j0 = tj << 4;

    const int  l16 = lane & 15;
    const bool lo  = lane < 16;

    const float* gb = g + (size_t)b * D;
    const float giv = gb[i0 + l16];
    const float gjv = gb[j0 + l16];

    v2f a  = { lo ? giv : 0.0f, 0.0f };
    v2f bm = { lo ? gjv : 0.0f, 0.0f };
    v8f c  = {};
    c = __builtin_amdgcn_wmma_f32_16x16x4_f32(
            false, a, false, bm, (short)0, c, false, false);

    const int rbase = lo ? 0 : 8;
    size_t off = (size_t)b * D * D + (size_t)(i0 + rbase) * D + (size_t)(j0 + l16);
    const float* sp = sigma + off;
    float*       op = out + off;

#pragma unroll
    for (int r = 0; r < 8; ++r) {
        float s = __builtin_nontemporal_load(sp);
        __builtin_nontemporal_store(s * c[r], op);
        sp += D;
        op += D;
    }
}

extern "C" void kernel_launch(void* const* d_in, const int* in_sizes, int n_in,
                              void* d_out, int out_size, void* d_ws, size_t ws_size,
                              hipStream_t stream) {
    const float* mu    = (const float*)d_in[0];
    const float* sigma = (const float*)d_in[1];
    float* out = (float*)d_out;
    float* g   = (float*)d_ws;

    const int BD  = in_sizes[0];            // B*D (131072)
    const int BDD = in_sizes[1];            // B*D*D (2^27, fits int)
    const int D   = BDD / BD;               // 1024
    const int B   = BD / D;                 // 128

    // Pass 1: mu_g -> out[0..BD), g -> workspace
    {
        int threads = 256;
        int blocks  = (BD + threads - 1) / threads;
        selu_fwd_g_kernel<<<blocks, threads, 0, stream>>>(mu, out, g, BD);
    }

    // Pass 2: sigma_g -> out[BD ..)
    {
        const int nj          = D >> 4;
        const int tiles_per_b = nj * nj;
        const int total_tiles = B * tiles_per_b;    // 524288 wave-tiles for 128x1024
        const int waves_per_block = 8;              // 256 threads
        const int blocks = (total_tiles + waves_per_block - 1) / waves_per_block;

        if (D == 1024) {
            sigma_outer_scale_kernel<1024><<<blocks, 256, 0, stream>>>(
                sigma, g, out + BD, total_tiles);
        } else if (D == 512) {
            sigma_outer_scale_kernel<512><<<blocks, 256, 0, stream>>>(
                sigma, g, out + BD, total_tiles);
        } else {
            sigma_outer_scale_generic<<<blocks, 256, 0, stream>>>(
                sigma, g, out + BD, D, nj, tiles_per_b, total_tiles);
        }
    }
}